// ARC2_65249143160996
// MI455X (gfx1250) — compile-verified
//
#include <hip/hip_runtime.h>

typedef float v2f __attribute__((ext_vector_type(2)));
typedef float v8f __attribute__((ext_vector_type(8)));

// Branch-free SELU: scale*max(x,0) + scale*alpha*(exp(min(x,0))-1).
// Uses the hardware v_exp_f32 path (__expf) -> ~6 VALU ops, no EXEC divergence.
__device__ __forceinline__ float selu_f(float x) {
    const float scale  = 1.0507009873554805f;
    const float salpha = 1.7580993408473766f;   // scale * alpha
    float e = __expf(fminf(x, 0.0f)) - 1.0f;    // 0 for x>0
    return fmaxf(x, 0.0f) * scale + salpha * e;
}

struct PhaseW { const float* W[5]; const float* b[5]; };

// Register-resident B-matrix fragments + bias for one layer.
// B layout mirrors A's K layout: vgpr j holds K=4t+j (lanes 0-15) / K=4t+j+2
// (lanes 16-31), N = lane%16 columns.
template<int K, int NOUT>
struct BFrag {
    static constexpr int KT = (K + 3) / 4;
    static constexpr int NT = (NOUT + 15) / 16;
    v2f   w[NT * KT];
    float bias[NT];
};

template<int K, int NOUT>
__device__ __forceinline__ void preloadB(const float* __restrict__ W,
                                         const float* __restrict__ Bv,
                                         BFrag<K, NOUT>& f)
{
    const int lane = threadIdx.x & 31;
    const int half = lane >> 4;
    const int l16  = lane & 15;
#pragma unroll
    for (int nt = 0; nt < BFrag<K, NOUT>::NT; ++nt) {
        const int ncol = nt * 16 + l16;
        const int ncc  = (ncol < NOUT) ? ncol : 0;
#pragma unroll
        for (int t = 0; t < BFrag<K, NOUT>::KT; ++t) {
            const int k0 = 4 * t + 2 * half;
            const int k1 = k0 + 1;
            const int ka = (k0 < K) ? k0 : 0;
            const int kb = (k1 < K) ? k1 : 0;
            float wx = W[ka * NOUT + ncc];
            float wy = W[kb * NOUT + ncc];
            v2f b;
            b.x = (k0 < K && ncol < NOUT) ? wx : 0.0f;
            b.y = (k1 < K && ncol < NOUT) ? wy : 0.0f;
            f.w[nt * BFrag<K, NOUT>::KT + t] = b;
        }
        f.bias[nt] = (ncol < NOUT) ? Bv[ncol] : 0.0f;
    }
}

// One 16-row MLP layer via V_WMMA_F32_16X16X4_F32 with register weights.
// A: M=lane%16 rows from LDS, K split {0,1}/{2,3} across lane halves.
// C/D: v8f, M across elements (+8 for upper lane half), N=lane%16.
template<bool ACT, int K, int NOUT>
__device__ __forceinline__ void layer16r(const float* __restrict__ in, int inS,
                                         const BFrag<K, NOUT>& f,
                                         float* __restrict__ out, int outS)
{
    const int lane = threadIdx.x & 31;
    const int half = lane >> 4;
    const int l16  = lane & 15;
    constexpr int KT = BFrag<K, NOUT>::KT;
    constexpr int NT = BFrag<K, NOUT>::NT;
#pragma unroll
    for (int nt = 0; nt < NT; ++nt) {
        const int ncol = nt * 16 + l16;
        v8f acc = {0.f, 0.f, 0.f, 0.f, 0.f, 0.f, 0.f, 0.f};
#pragma unroll
        for (int t = 0; t < KT; ++t) {
            const int k0 = 4 * t + 2 * half;
            const int k1 = k0 + 1;
            const int ka = (k0 < K) ? k0 : 0;
            const int kb = (k1 < K) ? k1 : 0;
            v2f a;
            float ax = in[l16 * inS + ka];
            float ay = in[l16 * inS + kb];
            a.x = (k0 < K) ? ax : 0.0f;
            a.y = (k1 < K) ? ay : 0.0f;
            acc = __builtin_amdgcn_wmma_f32_16x16x4_f32(false, a, false,
                                                        f.w[nt * KT + t],
                                                        (short)0, acc, false, false);
        }
        if (ncol < NOUT) {
            const float bias = f.bias[nt];
#pragma unroll
            for (int i = 0; i < 8; ++i) {
                float v = acc[i] + bias;
                if (ACT) v = selu_f(v);
                out[(i + 8 * half) * outS + ncol] = v;
            }
        }
    }
}

#define NW 8                     // waves per block (256 threads)
#define PWAVE (320 + 576 + 576)  // P(16x20) + T0(16x36) + T1(16x36) floats

__global__ void ARC2_zero_kernel(float* __restrict__ p, int n) {
    int i = blockIdx.x * blockDim.x + threadIdx.x;
    if (i < n) p[i] = 0.0f;
}

__global__ void __launch_bounds__(256)
ARC2_edge_kernel(const float* __restrict__ nodes, const float* __restrict__ edges,
                 const int* __restrict__ receivers, const int* __restrict__ senders,
                 PhaseW pm, PhaseW um, float* __restrict__ seg, long long E)
{
    __shared__ float smem[NW * PWAVE];
    const int wave = threadIdx.x >> 5;
    const int lane = threadIdx.x & 31;
    float* P  = smem + wave * PWAVE;   // 16x20: [edge(3) | h1_nodes(16)]
    float* T0 = P + 320;               // 16x36 ping
    float* T1 = T0 + 576;              // 16x36 pong

    // all weights register-resident for the whole kernel (one-time L2 reads)
    BFrag<6, 16>  fp0; preloadB(pm.W[0], pm.b[0], fp0);
    BFrag<16, 32> fp1; preloadB(pm.W[1], pm.b[1], fp1);
    BFrag<32, 32> fp2; preloadB(pm.W[2], pm.b[2], fp2);
    BFrag<32, 32> fp3; preloadB(pm.W[3], pm.b[3], fp3);
    BFrag<32, 16> fp4; preloadB(pm.W[4], pm.b[4], fp4);
    BFrag<19, 12> fu0; preloadB(um.W[0], um.b[0], fu0);
    BFrag<12, 12> fu1; preloadB(um.W[1], um.b[1], fu1);
    BFrag<12, 12> fu2; preloadB(um.W[2], um.b[2], fu2);
    BFrag<12, 8>  fu3; preloadB(um.W[3], um.b[3], fu3);
    BFrag<8, 3>   fu4; preloadB(um.W[4], um.b[4], fu4);

    const long long tiles = (E + 15) >> 4;
    const long long wstep = (long long)gridDim.x * NW;
    for (long long tile = (long long)blockIdx.x * NW + wave; tile < tiles; tile += wstep) {
        const long long e0 = tile << 4;
        int rcv = -1;
        if (lane < 16) {
            const long long e = e0 + lane;
            if (e < E) {
                rcv = receivers[e];
                const int snd = senders[e];
                T0[lane*36+0] = nodes[rcv*3+0];
                T0[lane*36+1] = nodes[rcv*3+1];
                T0[lane*36+2] = nodes[rcv*3+2];
                T0[lane*36+3] = nodes[snd*3+0];
                T0[lane*36+4] = nodes[snd*3+1];
                T0[lane*36+5] = nodes[snd*3+2];
                P[lane*20+0] = edges[e*3+0];
                P[lane*20+1] = edges[e*3+1];
                P[lane*20+2] = edges[e*3+2];
            } else {
#pragma unroll
                for (int c = 0; c < 6; ++c) T0[lane*36+c] = 0.0f;
                P[lane*20+0] = 0.f; P[lane*20+1] = 0.f; P[lane*20+2] = 0.f;
            }
        }
        // perm_nodes: 6->16->32->32->32->16 (SELU between) -> P cols 3..18
        layer16r<true >(T0, 36, fp0, T1, 36);
        layer16r<true >(T1, 36, fp1, T0, 36);
        layer16r<true >(T0, 36, fp2, T1, 36);
        layer16r<true >(T1, 36, fp3, T0, 36);
        layer16r<false>(T0, 36, fp4, P + 3, 20);
        // 10x upd_edges on concat([h_edges, h1_nodes]) = P cols 0..18
        for (int it = 0; it < 10; ++it) {
            layer16r<true >(P,  20, fu0, T0, 36);
            layer16r<true >(T0, 36, fu1, T1, 36);
            layer16r<true >(T1, 36, fu2, T0, 36);
            layer16r<true >(T0, 36, fu3, T1, 36);
            layer16r<false>(T1, 36, fu4, P,  20);
        }
        if (lane < 16) {
            const long long e = e0 + lane;
            if (e < E) {
                atomicAdd(&seg[rcv*3+0], P[lane*20+0]);
                atomicAdd(&seg[rcv*3+1], P[lane*20+1]);
                atomicAdd(&seg[rcv*3+2], P[lane*20+2]);
            }
        }
    }
}

__global__ void __launch_bounds__(256)
ARC2_node_kernel(const float* __restrict__ nodes, const float* __restrict__ seg,
                 const int* __restrict__ node_graph,
                 PhaseW pm, PhaseW um, float* __restrict__ glob, int N)
{
    __shared__ float smem[NW * PWAVE];
    const int wave = threadIdx.x >> 5;
    const int lane = threadIdx.x & 31;
    float* P  = smem + wave * PWAVE;   // 16x20: [h2_nodes(3) | h2_edges(16)]
    float* T0 = P + 320;
    float* T1 = T0 + 576;

    BFrag<3, 16>  fq0; preloadB(pm.W[0], pm.b[0], fq0);
    BFrag<16, 32> fq1; preloadB(pm.W[1], pm.b[1], fq1);
    BFrag<32, 32> fq2; preloadB(pm.W[2], pm.b[2], fq2);
    BFrag<32, 32> fq3; preloadB(pm.W[3], pm.b[3], fq3);
    BFrag<32, 16> fq4; preloadB(pm.W[4], pm.b[4], fq4);
    BFrag<19, 12> fv0; preloadB(um.W[0], um.b[0], fv0);
    BFrag<12, 12> fv1; preloadB(um.W[1], um.b[1], fv1);
    BFrag<12, 12> fv2; preloadB(um.W[2], um.b[2], fv2);
    BFrag<12, 8>  fv3; preloadB(um.W[3], um.b[3], fv3);
    BFrag<8, 3>   fv4; preloadB(um.W[4], um.b[4], fv4);

    const int tiles = (N + 15) >> 4;
    const int wstep = gridDim.x * NW;
    for (int tile = blockIdx.x * NW + wave; tile < tiles; tile += wstep) {
        const int n0 = tile << 4;
        int g = -1;
        if (lane < 16) {
            const int n = n0 + lane;
            if (n < N) {
                g = node_graph[n];
                T0[lane*36+0] = seg[n*3+0];
                T0[lane*36+1] = seg[n*3+1];
                T0[lane*36+2] = seg[n*3+2];
                P[lane*20+0] = nodes[n*3+0];
                P[lane*20+1] = nodes[n*3+1];
                P[lane*20+2] = nodes[n*3+2];
            } else {
                T0[lane*36+0] = 0.f; T0[lane*36+1] = 0.f; T0[lane*36+2] = 0.f;
                P[lane*20+0] = 0.f;  P[lane*20+1] = 0.f;  P[lane*20+2] = 0.f;
            }
        }
        // perm_edges: 3->16->32->32->32->16 -> P cols 3..18
        layer16r<true >(T0, 36, fq0, T1, 36);
        layer16r<true >(T1, 36, fq1, T0, 36);
        layer16r<true >(T0, 36, fq2, T1, 36);
        layer16r<true >(T1, 36, fq3, T0, 36);
        layer16r<false>(T0, 36, fq4, P + 3, 20);
        for (int it = 0; it < 10; ++it) {
            layer16r<true >(P,  20, fv0, T0, 36);
            layer16r<true >(T0, 36, fv1, T1, 36);
            layer16r<true >(T1, 36, fv2, T0, 36);
            layer16r<true >(T0, 36, fv3, T1, 36);
            layer16r<false>(T1, 36, fv4, P,  20);
        }
        // segment-sum concat([h2_nodes, h2_edges]) (= P cols 0..18) into globals.
        // node_graph is sorted -> whole tile usually maps to one graph: pre-reduce.
        const int g0 = __shfl(g, 0);
        const bool ok = (lane < 16) ? (g == g0 && g >= 0) : true;
        if (__all(ok)) {
            if (lane < 19) {
                float s = 0.0f;
#pragma unroll
                for (int r = 0; r < 16; ++r) s += P[r*20 + lane];
                atomicAdd(&glob[g0*19 + lane], s);
            }
        } else if (lane < 16) {
            const int n = n0 + lane;
            if (n < N) {
                for (int c = 0; c < 19; ++c)
                    atomicAdd(&glob[g*19 + c], P[lane*20 + c]);
            }
        }
    }
}

// tiny head: [B,19] -> [B,9]; negligible FLOPs, scalar fp32 is fine
__global__ void ARC2_predict_kernel(const float* __restrict__ glob, PhaseW w,
                                    float* __restrict__ out, int B)
{
    const int r = blockIdx.x * blockDim.x + threadIdx.x;
    if (r >= B) return;
    float h0[19], h1[12], h2[12], h3[12], h4[8];
    for (int c = 0; c < 19; ++c) h0[c] = glob[r*19 + c];
    for (int j = 0; j < 12; ++j) { float s = w.b[0][j];
        for (int k = 0; k < 19; ++k) s += h0[k] * w.W[0][k*12 + j]; h1[j] = selu_f(s); }
    for (int j = 0; j < 12; ++j) { float s = w.b[1][j];
        for (int k = 0; k < 12; ++k) s += h1[k] * w.W[1][k*12 + j]; h2[j] = selu_f(s); }
    for (int j = 0; j < 12; ++j) { float s = w.b[2][j];
        for (int k = 0; k < 12; ++k) s += h2[k] * w.W[2][k*12 + j]; h3[j] = selu_f(s); }
    for (int j = 0; j < 8; ++j)  { float s = w.b[3][j];
        for (int k = 0; k < 12; ++k) s += h3[k] * w.W[3][k*12 + j]; h4[j] = selu_f(s); }
    for (int j = 0; j < 9; ++j)  { float s = w.b[4][j];
        for (int k = 0; k < 8; ++k)  s += h4[k] * w.W[4][k*9 + j];  out[r*9 + j] = s; }
}

extern "C" void kernel_launch(void* const* d_in, const int* in_sizes, int n_in,
                              void* d_out, int out_size, void* d_ws, size_t ws_size,
                              hipStream_t stream)
{
    (void)n_in; (void)ws_size;
    const float* nodes = (const float*)d_in[0];
    const float* edges = (const float*)d_in[1];
    int idx = 2;
    PhaseW perm_nodes, upd_edges, perm_edges, upd_nodes, predict;
    PhaseW* ms[5] = { &perm_nodes, &upd_edges, &perm_edges, &upd_nodes, &predict };
    for (int m = 0; m < 5; ++m)
        for (int l = 0; l < 5; ++l) {
            ms[m]->W[l] = (const float*)d_in[idx++];
            ms[m]->b[l] = (const float*)d_in[idx++];
        }
    const int* receivers  = (const int*)d_in[52];
    const int* senders    = (const int*)d_in[53];
    const int* node_graph = (const int*)d_in[54];

    const int N = in_sizes[0] / 3;
    const long long E = (long long)in_sizes[1] / 3;
    const int B = out_size / 9;

    float* seg  = (float*)d_ws;            // [N,3]
    float* glob = seg + (size_t)N * 3;     // [B,19]
    float* out  = (float*)d_out;

    const int zn = N * 3 + B * 19;
    ARC2_zero_kernel<<<(zn + 255) / 256, 256, 0, stream>>>(seg, zn);

    const long long etiles = (E + 15) >> 4;
    long long egrid_ll = (etiles + NW - 1) / NW;
    if (egrid_ll > 4096) egrid_ll = 4096;
    ARC2_edge_kernel<<<(int)egrid_ll, 256, 0, stream>>>(
        nodes, edges, receivers, senders, perm_nodes, upd_edges, seg, E);

    const int ntiles = (N + 15) >> 4;
    int ngrid = (ntiles + NW - 1) / NW;
    if (ngrid > 4096) ngrid = 4096;
    ARC2_node_kernel<<<ngrid, 256, 0, stream>>>(
        nodes, seg, node_graph, perm_edges, upd_nodes, glob, N);

    ARC2_predict_kernel<<<(B + 63) / 64, 64, 0, stream>>>(glob, predict, out, B);
}